// NeuralJumpSDE_5935644803471
// MI455X (gfx1250) — compile-verified
//
#include <hip/hip_runtime.h>
#include <math.h>
#include <stdint.h>

#define D_LAT 128
#define D_HID 512
#define NSAMP 2048
#define NEV   64
#define ROWS  16
#define MAXS  512
#define CH    256          // hidden-column chunk
#define HSTR  260          // LDS stride for hidden chunks (260 % 64 == 4 -> bank-clean)

typedef __attribute__((ext_vector_type(2))) float v2f;
typedef __attribute__((ext_vector_type(8))) float v8f;
typedef __attribute__((ext_vector_type(4))) int   v4i;

typedef __attribute__((address_space(1))) v4i* as1_v4i_ptr;
typedef __attribute__((address_space(3))) v4i* as3_v4i_ptr;

__device__ __forceinline__ v8f wmma4(v2f a, v2f b, v8f c) {
  // D = A(16x4 f32) * B(4x16 f32) + C(16x16 f32)
  return __builtin_amdgcn_wmma_f32_16x16x4_f32(false, a, false, b, (short)0, c, false, false);
}

__device__ __forceinline__ float softplusf(float x) {
  return fmaxf(x, 0.0f) + log1pf(expf(-fabsf(x)));
}

// ---------------------------------------------------------------------------
// Schedule builder: bit-replicates host build_schedule (double precision).
// ---------------------------------------------------------------------------
__global__ void sched_k(const float* __restrict__ evt,
                        float* __restrict__ ts, float* __restrict__ dts,
                        float* __restrict__ evf, int* __restrict__ evi) {
  if (threadIdx.x == 0 && blockIdx.x == 0) {
    double t = 0.0;
    const double T = 1.0, dt = 0.01;
    int i = 0, s = 0;
    while (t < T && s < MAXS) {
      double nt = t + dt;
      if (i < NEV && (double)evt[i] <= nt) {
        double et = (double)evt[i];
        ts[s] = (float)t; dts[s] = (float)fmax(et - t, 0.0);
        evf[s] = 1.0f; evi[s] = i;
        t = et; ++i;
      } else {
        ts[s] = (float)t; dts[s] = 0.01f;
        evf[s] = 0.0f; evi[s] = 0;
        t = nt;
      }
      ++s;
    }
  }
}

// marks_proj[e,d] = event_marks[e,:] @ W_in[:,d] + b_in[d]
__global__ void mproj_k(const float* __restrict__ em, const float* __restrict__ Win,
                        const float* __restrict__ bin, float* __restrict__ mp) {
  int idx = blockIdx.x * blockDim.x + threadIdx.x;
  if (idx >= NEV * D_LAT) return;
  int e = idx >> 7, d = idx & 127;
  float acc = bin[d];
  for (int k = 0; k < D_LAT; ++k) acc += em[e * D_LAT + k] * Win[k * D_LAT + d];
  mp[idx] = acc;
}

// eventH[e,n] = marks_proj[e,:] @ Wj1[128:256, n] + bj1[n]
__global__ void evh_k(const float* __restrict__ mp, const float* __restrict__ Wj1,
                      const float* __restrict__ bj1, float* __restrict__ eH) {
  int idx = blockIdx.x * blockDim.x + threadIdx.x;
  if (idx >= NEV * D_HID) return;
  int e = idx >> 9, n = idx & 511;
  float acc = bj1[n];
  for (int k = 0; k < D_LAT; ++k)
    acc += mp[e * D_LAT + k] * Wj1[(size_t)(D_LAT + k) * D_HID + n];
  eH[idx] = acc;
}

__global__ void init_k(const float* __restrict__ z0, float* __restrict__ traj0) {
  int idx = blockIdx.x * blockDim.x + threadIdx.x;
  if (idx < NSAMP * D_LAT) traj0[idx] = z0[idx & 127];
}

// Transpose the six weight blocks (K-major -> per-lane-contiguous) once.
// T[0..2]: 512x128 views of Wd1/Wg1/Wj1 rows 0..127 (T[n*128+k] = W[k*512+n])
// T[3..5]: 128x512 views of Wd2/Wg2/Wj2            (T[n*512+k] = W[k*128+n])
__global__ void tw_k(const float* __restrict__ Wd1, const float* __restrict__ Wg1,
                     const float* __restrict__ Wj1, const float* __restrict__ Wd2,
                     const float* __restrict__ Wg2, const float* __restrict__ Wj2,
                     float* __restrict__ T) {
  int idx = blockIdx.x * blockDim.x + threadIdx.x;
  if (idx >= 6 * 65536) return;
  int mi = idx >> 16, r = idx & 65535;
  if (mi < 3) {
    const float* W = (mi == 0) ? Wd1 : (mi == 1) ? Wg1 : Wj1;
    int n = r >> 7, k = r & 127;
    T[(size_t)mi * 65536 + (size_t)n * D_LAT + k] = W[(size_t)k * D_HID + n];
  } else {
    const float* W = (mi == 3) ? Wd2 : (mi == 4) ? Wg2 : Wj2;
    int n = r >> 9, k = r & 511;
    T[(size_t)mi * 65536 + (size_t)n * D_HID + k] = W[(size_t)k * D_LAT + n];
  }
}

// ---------------------------------------------------------------------------
// Fragment loaders (TB=1: transposed weights -> one b64 per fragment).
// ---------------------------------------------------------------------------
template <int TB>
__device__ __forceinline__ v2f loadB1(const float* __restrict__ W,
                                      const float* __restrict__ WT, int k0, int gcol) {
  v2f b;
  if (TB) {
    b = *(const v2f*)(WT + (size_t)gcol * D_LAT + k0);
  } else {
    b.x = W[(size_t)k0 * D_HID + gcol];
    b.y = W[(size_t)(k0 + 1) * D_HID + gcol];
  }
  return b;
}
template <int TB>
__device__ __forceinline__ v2f loadB2(const float* __restrict__ W,
                                      const float* __restrict__ WT, int gk0, int n) {
  v2f b;
  if (TB) {
    b = *(const v2f*)(WT + (size_t)n * D_HID + gk0);
  } else {
    b.x = W[(size_t)gk0 * D_LAT + n];
    b.y = W[(size_t)(gk0 + 1) * D_LAT + n];
  }
  return b;
}

// Fused drift+diffusion layer-1 for one 256-column chunk: shares A fragments,
// 4 independent WMMA chains. Wave w owns local column tiles {w, w+8}.
template <int TB>
__device__ __forceinline__ void layer1_pair(
    const float (*zt)[132], float (*hd)[HSTR], float (*hg)[HSTR],
    const float* __restrict__ Wd1, const float* __restrict__ Wd1T,
    const float* __restrict__ Wg1, const float* __restrict__ Wg1T,
    const float* __restrict__ cbd, const float* __restrict__ cbg,
    int chunk, int w, int hf, int m) {
  v8f aD0 = {}, aD1 = {}, aG0 = {}, aG1 = {};
  const int c0 = w * 16 + m, c1 = (w + 8) * 16 + m;
  const int g0 = chunk * CH + c0, g1 = chunk * CH + c1;
  for (int kk = 0; kk < D_LAT / 4; ++kk) {
    const int k0 = kk * 4 + 2 * hf;   // A: lanes0-15 K={0,1}, lanes16-31 K={2,3}
    v2f a; a.x = zt[m][k0]; a.y = zt[m][k0 + 1];
    v2f bd0 = loadB1<TB>(Wd1, Wd1T, k0, g0);
    v2f bg0 = loadB1<TB>(Wg1, Wg1T, k0, g0);
    v2f bd1 = loadB1<TB>(Wd1, Wd1T, k0, g1);
    v2f bg1 = loadB1<TB>(Wg1, Wg1T, k0, g1);
    aD0 = wmma4(a, bd0, aD0);
    aG0 = wmma4(a, bg0, aG0);
    aD1 = wmma4(a, bd1, aD1);
    aG1 = wmma4(a, bg1, aG1);
  }
  const float bD0 = cbd[g0], bD1 = cbd[g1], bG0 = cbg[g0], bG1 = cbg[g1];
#pragma unroll
  for (int v = 0; v < 8; ++v) {
    int lr = v + 8 * hf;               // C/D: M = v + 8*half, N = lane%16
    hd[lr][c0] = tanhf(aD0[v] + bD0);
    hd[lr][c1] = tanhf(aD1[v] + bD1);
    hg[lr][c0] = tanhf(aG0[v] + bG0);
    hg[lr][c1] = tanhf(aG1[v] + bG1);
  }
}

// Fused drift+diffusion layer-2 partial accumulation over one K-chunk.
template <int TB>
__device__ __forceinline__ void layer2_pair(
    const float (*hd)[HSTR], const float (*hg)[HSTR],
    const float* __restrict__ Wd2, const float* __restrict__ Wd2T,
    const float* __restrict__ Wg2, const float* __restrict__ Wg2T,
    int chunk, int w, int hf, int m, v8f& dacc, v8f& gacc) {
  const int n = w * 16 + m;
  v8f d0 = {}, d1 = {}, g0 = {}, g1 = {};
  for (int kk = 0; kk < CH / 8; ++kk) {
#pragma unroll
    for (int u = 0; u < 2; ++u) {
      const int kl = (kk * 2 + u) * 4 + 2 * hf;
      const int gk = chunk * CH + kl;
      v2f ad; ad.x = hd[m][kl]; ad.y = hd[m][kl + 1];
      v2f ag; ag.x = hg[m][kl]; ag.y = hg[m][kl + 1];
      v2f bd = loadB2<TB>(Wd2, Wd2T, gk, n);
      v2f bg = loadB2<TB>(Wg2, Wg2T, gk, n);
      if (u == 0) { d0 = wmma4(ad, bd, d0); g0 = wmma4(ag, bg, g0); }
      else        { d1 = wmma4(ad, bd, d1); g1 = wmma4(ag, bg, g1); }
    }
  }
  dacc = dacc + d0 + d1;
  gacc = gacc + g0 + g1;
}

template <int TB>
__device__ __forceinline__ void layer1_jump(
    const float (*zt)[132], float (*hd)[HSTR],
    const float* __restrict__ Wj1, const float* __restrict__ Wj1T,
    const float* __restrict__ cb, int chunk, int w, int hf, int m) {
  v8f a0 = {}, a1 = {};
  const int c0 = w * 16 + m, c1 = (w + 8) * 16 + m;
  const int g0 = chunk * CH + c0, g1 = chunk * CH + c1;
  for (int kk = 0; kk < D_LAT / 4; ++kk) {
    const int k0 = kk * 4 + 2 * hf;
    v2f a; a.x = zt[m][k0]; a.y = zt[m][k0 + 1];
    v2f b0 = loadB1<TB>(Wj1, Wj1T, k0, g0);
    v2f b1 = loadB1<TB>(Wj1, Wj1T, k0, g1);
    a0 = wmma4(a, b0, a0);
    a1 = wmma4(a, b1, a1);
  }
  const float bb0 = cb[g0], bb1 = cb[g1];
#pragma unroll
  for (int v = 0; v < 8; ++v) {
    int lr = v + 8 * hf;
    hd[lr][c0] = fmaxf(a0[v] + bb0, 0.0f);
    hd[lr][c1] = fmaxf(a1[v] + bb1, 0.0f);
  }
}

template <int TB>
__device__ __forceinline__ void layer2_jump(
    const float (*hd)[HSTR],
    const float* __restrict__ Wj2, const float* __restrict__ Wj2T,
    int chunk, int w, int hf, int m, v8f& jacc) {
  const int n = w * 16 + m;
  v8f t0 = {}, t1 = {};
  for (int kk = 0; kk < CH / 8; ++kk) {
#pragma unroll
    for (int u = 0; u < 2; ++u) {
      const int kl = (kk * 2 + u) * 4 + 2 * hf;
      const int gk = chunk * CH + kl;
      v2f a; a.x = hd[m][kl]; a.y = hd[m][kl + 1];
      v2f b = loadB2<TB>(Wj2, Wj2T, gk, n);
      if (u == 0) t0 = wmma4(a, b, t0); else t1 = wmma4(a, b, t1);
    }
  }
  jacc = jacc + t0 + t1;
}

// ---------------------------------------------------------------------------
// One Euler-Maruyama (+ optional jump) step for a 16-row batch slab.
// ---------------------------------------------------------------------------
template <int TB>
__global__ __launch_bounds__(256) void step_k(
    const float* __restrict__ traj_cur, float* __restrict__ traj_next,
    float* __restrict__ out_events, const float* __restrict__ noise_s,
    const float* __restrict__ Wd1, const float* __restrict__ bd1,
    const float* __restrict__ Wd2, const float* __restrict__ bd2,
    const float* __restrict__ Wg1, const float* __restrict__ bg1,
    const float* __restrict__ Wg2, const float* __restrict__ bg2,
    const float* __restrict__ Wj1, const float* __restrict__ Wj2,
    const float* __restrict__ bj2, const float* __restrict__ eH,
    const float* __restrict__ Wd1T, const float* __restrict__ Wg1T,
    const float* __restrict__ Wj1T, const float* __restrict__ Wd2T,
    const float* __restrict__ Wg2T, const float* __restrict__ Wj2T,
    const float* __restrict__ jsp,
    const float* __restrict__ p_ts, const float* __restrict__ p_dt,
    const float* __restrict__ p_ev, const int* __restrict__ p_ei) {
  __shared__ float zt[ROWS][132];     // z / z_pre slab (cols 0..127)
  __shared__ float hd[ROWS][HSTR];    // drift-hidden chunk
  __shared__ float hg[ROWS][HSTR];    // diffusion-hidden chunk
  __shared__ float cbd[D_HID];        // drift layer-1 column bias
  __shared__ float cbg[D_HID];        // diffusion layer-1 column bias

  const int tid = threadIdx.x;
  const int lane = tid & 31;
  const int w = tid >> 5;
  const int hf = lane >> 4;
  const int m = lane & 15;
  const int rowBase = blockIdx.x * ROWS;

  const float tS = *p_ts;
  const float dtv = *p_dt;
  const bool ev = (*p_ev) > 0.5f;
  const int eidx = *p_ei;
  const float sq = sqrtf(dtv);
  const float js = *jsp;

  // ---- stage z slab into LDS (async-to-LDS when the toolchain exposes it) ----
#if defined(__gfx1250__) && __has_builtin(__builtin_amdgcn_global_load_async_to_lds_b128) && __has_builtin(__builtin_amdgcn_s_wait_asynccnt)
  for (int idx = tid; idx < ROWS * D_LAT / 4; idx += 256) {
    const int r = idx >> 5;
    const int c = (idx & 31) * 4;
    const float* gsrc = traj_cur + (size_t)(rowBase + r) * D_LAT + c;
    __builtin_amdgcn_global_load_async_to_lds_b128(
        (as1_v4i_ptr)(uintptr_t)gsrc,
        (as3_v4i_ptr)(unsigned int)(uintptr_t)&zt[r][c],
        0, 0);
  }
  __builtin_amdgcn_s_wait_asynccnt(0);
#else
  for (int idx = tid; idx < ROWS * D_LAT; idx += 256) {
    int r = idx >> 7, c = idx & 127;
    zt[r][c] = traj_cur[(size_t)(rowBase + r) * D_LAT + c];
  }
#endif

  // layer-1 biases: t*W[128,:] + b  (time-column folded into a column bias)
  for (int idx = tid; idx < D_HID; idx += 256) {
    cbd[idx] = tS * Wd1[(size_t)D_LAT * D_HID + idx] + bd1[idx];
    cbg[idx] = tS * Wg1[(size_t)D_LAT * D_HID + idx] + bg1[idx];
  }
  __syncthreads();

  v8f dacc = {}, gacc = {};
  for (int ch = 0; ch < 2; ++ch) {
    layer1_pair<TB>(zt, hd, hg, Wd1, Wd1T, Wg1, Wg1T, cbd, cbg, ch, w, hf, m);
    __syncthreads();
    layer2_pair<TB>(hd, hg, Wd2, Wd2T, Wg2, Wg2T, ch, w, hf, m, dacc, gacc);
    __syncthreads();
  }

  // ---- Euler-Maruyama combine (wave w owns cols w*16..w*16+15, all 16 rows) ----
  {
    const int n = w * 16 + m;
    const float bd2n = bd2[n], bg2n = bg2[n];
#pragma unroll
    for (int v = 0; v < 8; ++v) {
      int lr = v + 8 * hf;
      int gr = rowBase + lr;
      float zv = zt[lr][n];
      float dr = dacc[v] + bd2n;
      float df = softplusf(gacc[v] + bg2n);
      float nz = noise_s[(size_t)gr * D_LAT + n];
      float zp = zv + dr * dtv + df * nz * sq;
      zt[lr][n] = zp;                                   // z_pre replaces z
      if (ev) out_events[((size_t)eidx * NSAMP + gr) * D_LAT + n] = zp;
      else    traj_next[(size_t)gr * D_LAT + n] = zp;   // z_new == z_pre
    }
  }
  __syncthreads();

  if (ev) {
    // jump = js * tanh(relu(z_pre@Wj1[0:128] + eventH[eidx]) @ Wj2 + bj2)
    for (int idx = tid; idx < D_HID; idx += 256)
      cbd[idx] = eH[(size_t)eidx * D_HID + idx];
    __syncthreads();
    v8f jacc = {};
    for (int ch = 0; ch < 2; ++ch) {
      layer1_jump<TB>(zt, hd, Wj1, Wj1T, cbd, ch, w, hf, m);
      __syncthreads();
      layer2_jump<TB>(hd, Wj2, Wj2T, ch, w, hf, m, jacc);
      __syncthreads();
    }
    const int n = w * 16 + m;
    const float bj2n = bj2[n];
#pragma unroll
    for (int v = 0; v < 8; ++v) {
      int lr = v + 8 * hf;
      int gr = rowBase + lr;
      float jump = js * tanhf(jacc[v] + bj2n);
      traj_next[(size_t)gr * D_LAT + n] = zt[lr][n] + jump;
    }
  }
}

// ---------------------------------------------------------------------------
extern "C" void kernel_launch(void* const* d_in, const int* in_sizes, int n_in,
                              void* d_out, int out_size, void* d_ws, size_t ws_size,
                              hipStream_t stream) {
  const float* evt = (const float*)d_in[0];
  const float* em  = (const float*)d_in[1];
  const float* noise = (const float*)d_in[2];
  const float* Win = (const float*)d_in[3];
  const float* bin = (const float*)d_in[4];
  const float* Wd1 = (const float*)d_in[5];
  const float* bd1 = (const float*)d_in[6];
  const float* Wd2 = (const float*)d_in[7];
  const float* bd2 = (const float*)d_in[8];
  const float* Wg1 = (const float*)d_in[9];
  const float* bg1 = (const float*)d_in[10];
  const float* Wg2 = (const float*)d_in[11];
  const float* bg2 = (const float*)d_in[12];
  const float* Wj1 = (const float*)d_in[13];
  const float* bj1 = (const float*)d_in[14];
  const float* Wj2 = (const float*)d_in[15];
  const float* bj2 = (const float*)d_in[16];
  const float* js  = (const float*)d_in[17];
  const float* z0  = (const float*)d_in[18];
  (void)in_sizes; (void)n_in;

  const long long per = (long long)NSAMP * D_LAT;
  int S = (int)((long long)out_size / per) - 1 - NEV;   // out = events[64] ++ traj[S+1]
  if (S < 1) S = 1;
  if (S > MAXS) S = MAXS;

  float* ws  = (float*)d_ws;
  float* ts  = ws;                 // [512]
  float* dts = ws + 512;           // [512]
  float* evf = ws + 1024;          // [512]
  int*   evi = (int*)(ws + 1536);  // [512]
  float* mp  = ws + 2048;          // [64*128]
  float* eH  = mp + NEV * D_LAT;   // [64*512]  -> ends at ws + 43008
  float* TW  = ws + 43008;         // [6*65536] transposed weights (optional)

  const size_t needT = (size_t)(43008 + 6 * 65536) * sizeof(float);
  const int useT = (ws_size >= needT) ? 1 : 0;
  float* Wd1T = TW;
  float* Wg1T = TW + 1 * 65536;
  float* Wj1T = TW + 2 * 65536;
  float* Wd2T = TW + 3 * 65536;
  float* Wg2T = TW + 4 * 65536;
  float* Wj2T = TW + 5 * 65536;

  float* out_events = (float*)d_out;
  float* traj = out_events + (size_t)NEV * per;

  sched_k<<<1, 32, 0, stream>>>(evt, ts, dts, evf, evi);
  mproj_k<<<(NEV * D_LAT + 255) / 256, 256, 0, stream>>>(em, Win, bin, mp);
  evh_k<<<(NEV * D_HID + 255) / 256, 256, 0, stream>>>(mp, Wj1, bj1, eH);
  init_k<<<(NSAMP * D_LAT + 255) / 256, 256, 0, stream>>>(z0, traj);
  if (useT)
    tw_k<<<(6 * 65536 + 255) / 256, 256, 0, stream>>>(Wd1, Wg1, Wj1, Wd2, Wg2, Wj2, TW);

  for (int s = 0; s < S; ++s) {
    if (useT) {
      step_k<1><<<NSAMP / ROWS, 256, 0, stream>>>(
          traj + (size_t)s * per, traj + (size_t)(s + 1) * per, out_events,
          noise + (size_t)s * per,
          Wd1, bd1, Wd2, bd2, Wg1, bg1, Wg2, bg2, Wj1, Wj2, bj2, eH,
          Wd1T, Wg1T, Wj1T, Wd2T, Wg2T, Wj2T, js,
          ts + s, dts + s, evf + s, evi + s);
    } else {
      step_k<0><<<NSAMP / ROWS, 256, 0, stream>>>(
          traj + (size_t)s * per, traj + (size_t)(s + 1) * per, out_events,
          noise + (size_t)s * per,
          Wd1, bd1, Wd2, bd2, Wg1, bg1, Wg2, bg2, Wj1, Wj2, bj2, eH,
          Wd1T, Wg1T, Wj1T, Wd2T, Wg2T, Wj2T, js,
          ts + s, dts + s, evf + s, evi + s);
    }
  }
}